// BCEAndMultiLabelMarginLoss_18880676233272
// MI455X (gfx1250) — compile-verified
//
#include <hip/hip_runtime.h>

// Problem constants (B=16, T=1024, V=128)
#define VDIM 128
#define W_BCE 0.7f
#define W_MLM 0.3f

typedef __attribute__((ext_vector_type(2))) float v2f;
typedef __attribute__((ext_vector_type(8))) float v8f;

// Wave32 sum of per-lane scalars using V_WMMA_F32_16X16X4_F32 with an
// all-ones B matrix. A-layout (ISA 7.12.2, 32-bit A 16x4): lanes 0-15 hold
// K=0 (VGPR0) / K=1 (VGPR1); lanes 16-31 hold K=2/K=3; M = lane%16.
// Setting a.x = s_lane, a.y = 0 gives A[m][0]=s_m, A[m][2]=s_{m+16}.
// With B = ones: D[m][n] = s_m + s_{m+16}. Summing the 8 D VGPRs in a lane
// gives partial sums over 8 consecutive M; lane 0 covers M=0..7, lane 16
// covers M=8..15, so readlane(0) + readlane(16) = full 32-lane sum.
__device__ __forceinline__ float wave_sum_wmma(float s) {
#if __has_builtin(__builtin_amdgcn_wmma_f32_16x16x4_f32)
    v2f a; a[0] = s;    a[1] = 0.0f;
    v2f b; b[0] = 1.0f; b[1] = 1.0f;
    v8f c = {};
    v8f d = __builtin_amdgcn_wmma_f32_16x16x4_f32(
        /*neg_a=*/false, a, /*neg_b=*/false, b,
        /*c_mod=*/(short)0, c, /*reuse_a=*/false, /*reuse_b=*/false);
    float ls = ((d[0] + d[1]) + (d[2] + d[3])) + ((d[4] + d[5]) + (d[6] + d[7]));
    float lo = __int_as_float(__builtin_amdgcn_readlane(__float_as_int(ls), 0));
    float hi = __int_as_float(__builtin_amdgcn_readlane(__float_as_int(ls), 16));
    return lo + hi;
#else
    // Fallback: log2(32)-step shuffle reduction.
    for (int off = 16; off > 0; off >>= 1) s += __shfl_down(s, off, 32);
    return __int_as_float(__builtin_amdgcn_readlane(__float_as_int(s), 0));
#endif
}

// One row (V=128 fp32) per wave32: lane L owns elements [4L, 4L+3] via a
// single 16B load per operand (global_load_b128, 512B contiguous per wave).
// BCE is elementwise; the multilabel-margin term iterates only over actual
// positive classes (ballot -> readlane broadcast), ~4 per row, so pairwise
// work is O(#pos * V) instead of O(V^2). Everything accumulates into a
// single per-lane scalar; final scaling by 1/(N*V) happens in finalize.
__global__ void __launch_bounds__(256)
loss_partial_kernel(const float* __restrict__ logits,
                    const float* __restrict__ targets,
                    float* __restrict__ partials,
                    int nrows) {
    const int lane          = threadIdx.x & 31;
    const int waveInBlock   = threadIdx.x >> 5;
    const int wavesPerBlock = blockDim.x >> 5;
    const int waveId        = blockIdx.x * wavesPerBlock + waveInBlock;
    const int nWaves        = gridDim.x * wavesPerBlock;

    float acc = 0.0f;

    for (int row = waveId; row < nrows; row += nWaves) {
        const float4 xv = reinterpret_cast<const float4*>(logits  + (size_t)row * VDIM)[lane];
        const float4 tv = reinterpret_cast<const float4*>(targets + (size_t)row * VDIM)[lane];
        float xs[4] = {xv.x, xv.y, xv.z, xv.w};
        float ts[4] = {tv.x, tv.y, tv.z, tv.w};

        // --- BCE-with-logits, numerically stable form
        float bce = 0.0f;
        bool  pos[4];
#pragma unroll
        for (int s = 0; s < 4; ++s) {
            float x = xs[s];
            bce += fmaxf(x, 0.0f) - x * ts[s] + __logf(1.0f + __expf(-fabsf(x)));
            pos[s] = ts[s] > 0.5f;
        }

        // --- MultiLabelMargin: sum over (positive p, negative n) of
        //     relu(1 - x[p] + x[n]); enumerate positives via wave ballot.
        float hinge = 0.0f;
#pragma unroll
        for (int s = 0; s < 4; ++s) {
            unsigned m = __builtin_amdgcn_ballot_w32(pos[s]);
            while (m) {
                int src = __builtin_ctz(m);
                m &= m - 1;
                float xp = __int_as_float(
                    __builtin_amdgcn_readlane(__float_as_int(xs[s]), src));
                float cc = 1.0f - xp;
#pragma unroll
                for (int n = 0; n < 4; ++n) {
                    float h = fmaxf(cc + xs[n], 0.0f);
                    hinge += pos[n] ? 0.0f : h;
                }
            }
        }

        acc += W_BCE * bce + W_MLM * hinge;
    }

    // Cross-lane reduce via WMMA (EXEC is all-ones here: no divergent exits).
    float wsum = wave_sum_wmma(acc);

    __shared__ float smem[32];
    if (lane == 0) smem[waveInBlock] = wsum;
    __syncthreads();
    if (threadIdx.x == 0) {
        float b = 0.0f;
        for (int i = 0; i < wavesPerBlock; ++i) b += smem[i];
        partials[blockIdx.x] = b;
    }
}

__global__ void __launch_bounds__(256)
finalize_kernel(const float* __restrict__ partials, int n,
                float* __restrict__ out, float scale) {
    const int lane        = threadIdx.x & 31;
    const int waveInBlock = threadIdx.x >> 5;

    float s = 0.0f;
    for (int i = threadIdx.x; i < n; i += blockDim.x) s += partials[i];

    float wsum = wave_sum_wmma(s);

    __shared__ float smem[32];
    if (lane == 0) smem[waveInBlock] = wsum;
    __syncthreads();
    if (threadIdx.x == 0) {
        float b = 0.0f;
        const int wpb = blockDim.x >> 5;
        for (int i = 0; i < wpb; ++i) b += smem[i];
        out[0] = b * scale;
    }
}

extern "C" void kernel_launch(void* const* d_in, const int* in_sizes, int n_in,
                              void* d_out, int out_size, void* d_ws, size_t ws_size,
                              hipStream_t stream) {
    const float* logits  = (const float*)d_in[0];
    const float* targets = (const float*)d_in[1];

    const int total = in_sizes[0];       // B*T*V
    const int nrows = total / VDIM;      // B*T = 16384

    int nblocks = 2048;                  // 8 waves/block -> 1 row per wave
    size_t needed = (size_t)nblocks * sizeof(float);
    if (needed > ws_size) nblocks = (int)(ws_size / sizeof(float));
    if (nblocks < 1) nblocks = 1;

    float* partials = (float*)d_ws;

    loss_partial_kernel<<<nblocks, 256, 0, stream>>>(logits, targets, partials, nrows);

    const float scale = 1.0f / ((float)nrows * (float)VDIM);
    finalize_kernel<<<1, 256, 0, stream>>>(partials, nblocks, (float*)d_out, scale);
}